// GraphConvLayer_532575944843
// MI455X (gfx1250) — compile-verified
//
#include <hip/hip_runtime.h>
#include <hip/hip_bf16.h>

typedef __attribute__((ext_vector_type(2))) float v2f;
typedef __attribute__((ext_vector_type(8))) float v8f;

#define NNODES 100000
#define NEDGES 1600000
#define DFEAT  128
#define BN_EPS 1e-5f

// ---------------- workspace layout (in floats) ----------------
#define WS_AGG    0ull                                   // N*128
#define WS_CNT    (WS_AGG + (size_t)NNODES * DFEAT)      // N
#define WS_SUMS   (WS_CNT + (size_t)NNODES)              // 128
#define WS_SUMSQ  (WS_SUMS + 128ull)                     // 128
#define WS_SCALE  (WS_SUMSQ + 128ull)                    // 128
#define WS_SHIFT  (WS_SCALE + 128ull)                    // 128
#define WS_H      (WS_SHIFT + 128ull)                    // N*128
#define WS_ZERO_FLOATS ((size_t)NNODES * DFEAT + (size_t)NNODES + 256ull)

// ---------------------------------------------------------------
// Kernel 1: h = x @ W^T + b via V_WMMA_F32_16X16X4_F32.
// One wave computes one 16x16 tile of h. K = 128 -> 32 WMMA steps.
// A (16x4 f32): lane L holds row M = L%16; VGPR0 = K = k0 + 2*(L/16),
// VGPR1 = K+1.  B (4x16): lane L holds col N = L%16, same K split.
// C/D: VGPR r -> M = r + 8*(L/16), N = L%16.
// ---------------------------------------------------------------
__global__ __launch_bounds__(256) void gemm_wmma_f32(
    const float* __restrict__ x, const float* __restrict__ W,
    const float* __restrict__ bias, float* __restrict__ h) {
  const int wave = blockIdx.x * (blockDim.x >> 5) + (threadIdx.x >> 5);
  const int lane = threadIdx.x & 31;
  const int mTile = wave >> 3;          // 6250 row tiles
  const int nTile = wave & 7;           // 8 col tiles
  if (mTile * 16 >= NNODES) return;     // wave-uniform; EXEC stays all-ones

  const int m0 = mTile * 16;
  const int n0 = nTile * 16;
  const int l15 = lane & 15;
  const int khalf = (lane >> 4) * 2;    // 0 for lanes 0-15, 2 for 16-31

  const float* xrow = x + (size_t)(m0 + l15) * DFEAT + khalf;
  const float* wrow = W + (size_t)(n0 + l15) * DFEAT + khalf;

  v8f c;
  const float bv = bias[n0 + l15];
  #pragma unroll
  for (int i = 0; i < 8; ++i) c[i] = bv;

  #pragma unroll
  for (int k0 = 0; k0 < DFEAT; k0 += 4) {
    v2f a = *(const v2f*)(xrow + k0);
    v2f b = *(const v2f*)(wrow + k0);
    c = __builtin_amdgcn_wmma_f32_16x16x4_f32(
        /*neg_a=*/false, a, /*neg_b=*/false, b,
        /*c_mod=*/(short)0, c, /*reuse_a=*/false, /*reuse_b=*/false);
  }

  const int mr = m0 + ((lane >> 4) << 3);  // +8 for upper half-wave
  float* hout = h + (size_t)mr * DFEAT + (n0 + l15);
  #pragma unroll
  for (int r = 0; r < 8; ++r) hout[(size_t)r * DFEAT] = c[r];
}

// ---------------------------------------------------------------
// Kernel 2: edge scatter-add. One wave per edge; each lane moves a
// float4 (4 features) from h[col] into agg[row] via hardware f32
// atomics; lane 0 counts the edge.
// ---------------------------------------------------------------
__global__ __launch_bounds__(256) void edge_scatter(
    const long long* __restrict__ ei, const float* __restrict__ h,
    float* __restrict__ agg, float* __restrict__ cnt) {
  const long long gid  = (long long)blockIdx.x * blockDim.x + threadIdx.x;
  const long long edge = gid >> 5;
  const int lane = (int)(gid & 31);
  if (edge >= NEDGES) return;

  const long long row = ei[edge];
  const long long col = ei[(long long)NEDGES + edge];

  const float4 v = ((const float4*)(h + col * DFEAT))[lane];
  float* dst = agg + row * DFEAT + lane * 4;
  unsafeAtomicAdd(dst + 0, v.x);
  unsafeAtomicAdd(dst + 1, v.y);
  unsafeAtomicAdd(dst + 2, v.z);
  unsafeAtomicAdd(dst + 3, v.w);
  if (lane == 0) unsafeAtomicAdd(cnt + row, 1.0f);
}

// ---------------------------------------------------------------
// Kernel 3: hres = h + agg/(cnt+1e-8); accumulate per-feature
// sum / sumsq. Block = 128 threads (thread == feature), grid-stride
// over rows, 2 atomics per thread at the end.
// ---------------------------------------------------------------
__global__ __launch_bounds__(128) void residual_stats(
    const float* __restrict__ h, const float* __restrict__ agg,
    const float* __restrict__ cnt, float* __restrict__ hres,
    float* __restrict__ sums, float* __restrict__ sumsq) {
  const int f = threadIdx.x;
  float s = 0.0f, s2 = 0.0f;
  for (int row = blockIdx.x; row < NNODES; row += gridDim.x) {
    const float invc = 1.0f / (cnt[row] + 1e-8f);
    const size_t idx = (size_t)row * DFEAT + f;
    const float v = h[idx] + agg[idx] * invc;
    hres[idx] = v;
    s += v;
    s2 += v * v;
  }
  unsafeAtomicAdd(&sums[f], s);
  unsafeAtomicAdd(&sumsq[f], s2);
}

// ---------------------------------------------------------------
// Kernel 4: fold BN stats into scale/shift (128 threads, 1 block).
// ---------------------------------------------------------------
__global__ void bn_fold(const float* __restrict__ sums,
                        const float* __restrict__ sumsq,
                        const float* __restrict__ gamma,
                        const float* __restrict__ beta,
                        float* __restrict__ scale, float* __restrict__ shift) {
  const int f = threadIdx.x;
  const float invN = 1.0f / (float)NNODES;
  const float mean = sums[f] * invN;
  const float var = sumsq[f] * invN - mean * mean;  // biased, matches ref
  const float sc = rsqrtf(var + BN_EPS) * gamma[f];
  scale[f] = sc;
  shift[f] = beta[f] - mean * sc;
}

// ---------------------------------------------------------------
// Kernel 5: out = hres * scale[f] + shift[f], float4 streaming,
// in place on d_out.
// ---------------------------------------------------------------
__global__ __launch_bounds__(256) void bn_apply(
    float* __restrict__ out, const float* __restrict__ scale,
    const float* __restrict__ shift) {
  const size_t tid = (size_t)blockIdx.x * blockDim.x + threadIdx.x;
  const size_t total4 = (size_t)NNODES * DFEAT / 4;
  if (tid >= total4) return;
  const int f0 = (int)((tid * 4) & (DFEAT - 1));
  float4 v = ((float4*)out)[tid];
  v.x = v.x * scale[f0 + 0] + shift[f0 + 0];
  v.y = v.y * scale[f0 + 1] + shift[f0 + 1];
  v.z = v.z * scale[f0 + 2] + shift[f0 + 2];
  v.w = v.w * scale[f0 + 3] + shift[f0 + 3];
  ((float4*)out)[tid] = v;
}

extern "C" void kernel_launch(void* const* d_in, const int* in_sizes, int n_in,
                              void* d_out, int out_size, void* d_ws, size_t ws_size,
                              hipStream_t stream) {
  const float*     x     = (const float*)d_in[0];
  const long long* ei    = (const long long*)d_in[1];   // int64 edge_index (2,E)
  const float*     W     = (const float*)d_in[2];
  const float*     bias  = (const float*)d_in[3];
  const float*     gamma = (const float*)d_in[4];
  const float*     beta  = (const float*)d_in[5];

  float* ws    = (float*)d_ws;
  float* agg   = ws + WS_AGG;
  float* cnt   = ws + WS_CNT;
  float* sums  = ws + WS_SUMS;
  float* sumsq = ws + WS_SUMSQ;
  float* scale = ws + WS_SCALE;
  float* shift = ws + WS_SHIFT;
  float* h     = ws + WS_H;
  float* out   = (float*)d_out;

  // 1) zero accumulators (agg, cnt, sums, sumsq are contiguous)
  hipMemsetAsync(ws, 0, WS_ZERO_FLOATS * sizeof(float), stream);

  // 2) GEMM: (N/16)*(128/16) waves, 8 waves per 256-thread block
  const int tiles = (NNODES / 16) * (DFEAT / 16);     // 50000 waves
  gemm_wmma_f32<<<(tiles + 7) / 8, 256, 0, stream>>>(x, W, bias, h);

  // 3) scatter: 32 threads per edge
  const long long sthreads = (long long)NEDGES * 32;
  edge_scatter<<<(int)((sthreads + 255) / 256), 256, 0, stream>>>(ei, h, agg, cnt);

  // 4) residual + per-feature stats (hres -> d_out)
  residual_stats<<<2048, 128, 0, stream>>>(h, agg, cnt, out, sums, sumsq);

  // 5) fold stats
  bn_fold<<<1, 128, 0, stream>>>(sums, sumsq, gamma, beta, scale, shift);

  // 6) normalize in place
  const size_t total4 = (size_t)NNODES * DFEAT / 4;
  bn_apply<<<(int)((total4 + 255) / 256), 256, 0, stream>>>(out, scale, shift);
}